// v_LPTC_80255758893138
// MI455X (gfx1250) — compile-verified
//
#include <hip/hip_runtime.h>

// CDNA5 / gfx1250, wave32. Spin contraction on V_WMMA_F32_16X16X4_F32
// (K=4 == spin dim); color hops on VALU FMAs that co-execute with XDL.
// HBM-bound on the 302 MB single-use weight stream -> all 18 weight
// fragments are NT-loaded up front so the wave never stalls per-WMMA.

typedef __attribute__((ext_vector_type(2))) float v2f;
typedef __attribute__((ext_vector_type(8))) float v8f;

#define NSITE 65536   // 16^4
#define NPATH 9
#define FIN   2

static __device__ __forceinline__ v2f nt_load_v2(const float* p) {
  // w is 302 MB single-use (> 192 MB L2): stream it non-temporally.
  return __builtin_nontemporal_load((const v2f*)p);
}

__global__ __launch_bounds__(256) void lptc_wmma_kernel(
    const float* __restrict__ x_re, const float* __restrict__ x_im,
    const float* __restrict__ U_re, const float* __restrict__ U_im,
    const float* __restrict__ w_re, const float* __restrict__ w_im,
    float* __restrict__ out)
{
  const int lane = threadIdx.x & 31;
  const int wave = blockIdx.x * (blockDim.x >> 5) + (threadIdx.x >> 5);
  const int site = wave;                 // one wave per lattice site

  // --- lane roles ----------------------------------------------------------
  const int n    = lane & 15;            // A row m  AND  B/D column n
  const int half = lane >> 4;            // K half: this lane carries K=2h,2h+1
  const int toff = half << 1;

  // B/D column decode: n = c*2 + sigma (columns 0..5 active)
  const int  c      = n >> 1;
  const int  sigma  = n & 1;             // 0 = real part, 1 = imag part
  const bool active = (n < 6);
  const int  ccl    = active ? c : 0;    // clamped color: branchless loads

  // A row decode: m = o*8 + s*2 + rho
  const int o_a = n >> 3;
  const int s_a = (n >> 1) & 3;
  const int rho = n & 1;                 // 0 -> w_re, 1 -> w_im

  const int sx = (site >> 12) & 15, sy = (site >> 8) & 15,
            sz = (site >> 4) & 15,  st =  site        & 15;

  // --- phase 1: issue ALL 18 non-temporal weight loads ---------------------
  // (exactly-once pass over w; loads pipeline, waits amortize)
  const float* Wp = rho ? w_im : w_re;
  v2f a[FIN][NPATH];
  #pragma unroll
  for (int i = 0; i < FIN; ++i) {
    #pragma unroll
    for (int p = 0; p < NPATH; ++p) {
      const size_t woff =
          ((size_t)((i * 2 + o_a) * NPATH + p) * NSITE + (size_t)site) * 16
          + (size_t)(s_a * 4 + toff);
      a[i][p] = nt_load_v2(Wp + woff);
    }
  }

  // --- phase 2: hop + WMMA accumulate --------------------------------------
  v8f acc0 = {};   // fin = 0 contributions
  v8f acc1 = {};   // fin = 1 contributions (independent WMMA chain)

  #pragma unroll
  for (int p = 0; p < NPATH; ++p) {
    int nsite = site;
    float ur[3] = {0.f, 0.f, 0.f}, ui[3] = {0.f, 0.f, 0.f};

    if (p > 0) {
      const int mu  = (p - 1) >> 1;
      const int dir = (p & 1) ? 1 : -1;
      int cc[4] = {sx, sy, sz, st};
      cc[mu] = (cc[mu] + dir + 16) & 15;               // periodic hop
      nsite  = ((cc[0] * 16 + cc[1]) * 16 + cc[2]) * 16 + cc[3];
      const int usite = (dir > 0) ? site : nsite;      // U_mu(x) vs U_mu(x-mu)
      const float* Ur = U_re + ((size_t)mu * NSITE + usite) * 9;
      const float* Ui = U_im + ((size_t)mu * NSITE + usite) * 9;
      if (dir > 0) {                                   // row c of U
        #pragma unroll
        for (int b = 0; b < 3; ++b) { ur[b] = Ur[ccl * 3 + b]; ui[b] = Ui[ccl * 3 + b]; }
      } else {                                         // Ueff[c][b] = conj(U[b][c])
        #pragma unroll
        for (int b = 0; b < 3; ++b) { ur[b] = Ur[b * 3 + ccl]; ui[b] = -Ui[b * 3 + ccl]; }
      }
    }

    #pragma unroll
    for (int i = 0; i < FIN; ++i) {
      // B operand: T{sigma}[K = toff, toff+1] for color c (branchless)
      const float* Xr = x_re + ((size_t)i * NSITE + nsite) * 12;
      const float* Xi = x_im + ((size_t)i * NSITE + nsite) * 12;
      v2f bmat;
      #pragma unroll
      for (int t2 = 0; t2 < 2; ++t2) {
        const int t = toff + t2;
        float tr, ti;
        if (p == 0) {                    // identity path: no transport
          tr = Xr[t * 3 + ccl];
          ti = Xi[t * 3 + ccl];
        } else {                         // 3x3 complex color hop
          tr = 0.f; ti = 0.f;
          #pragma unroll
          for (int b = 0; b < 3; ++b) {
            const float xr = Xr[t * 3 + b], xi = Xi[t * 3 + b];
            tr = fmaf(ur[b], xr, fmaf(-ui[b], xi, tr));
            ti = fmaf(ur[b], xi, fmaf( ui[b], xr, ti));
          }
        }
        const float v = sigma ? ti : tr;
        bmat[t2] = active ? v : 0.f;     // cndmask, no exec branching
      }

      // D += A(16x4 f32) x B(4x16 f32); two independent accumulator chains
      if (i == 0) {
        acc0 = __builtin_amdgcn_wmma_f32_16x16x4_f32(
            false, a[0][p], false, bmat, (short)0, acc0, false, false);
      } else {
        acc1 = __builtin_amdgcn_wmma_f32_16x16x4_f32(
            false, a[1][p], false, bmat, (short)0, acc1, false, false);
      }
    }
  }

  const v8f acc = acc0 + acc1;           // sum over input features

  // --- recombine complex parts and store ----------------------------------
  // Lane (c,sigma) in half h holds D[8h + j, n] in acc[j]; 8h+j = o*8+s*2+rho.
  // out_re = D[rho0,sig0] - D[rho1,sig1];  out_im = D[rho0,sig1] + D[rho1,sig0]
  const int oo = half;                   // output feature from D-row half
  #pragma unroll
  for (int s = 0; s < 4; ++s) {
    const float mine    = acc[2 * s];                        // rho = 0
    const float partner = __shfl_xor(acc[2 * s + 1], 1, 32); // other sigma, rho=1
    const float val     = sigma ? (mine + partner) : (mine - partner);
    if (active) {
      const size_t off = (((size_t)oo * NSITE + site) * 4 + s) * 3 + c;
      out[2 * off + sigma] = val;        // complex64 interleaved re,im
    }
  }
}

extern "C" void kernel_launch(void* const* d_in, const int* in_sizes, int n_in,
                              void* d_out, int out_size, void* d_ws, size_t ws_size,
                              hipStream_t stream) {
  (void)in_sizes; (void)n_in; (void)d_ws; (void)ws_size; (void)out_size;
  const float* x_re = (const float*)d_in[0];
  const float* x_im = (const float*)d_in[1];
  const float* U_re = (const float*)d_in[2];
  const float* U_im = (const float*)d_in[3];
  const float* w_re = (const float*)d_in[4];
  const float* w_im = (const float*)d_in[5];
  float* out = (float*)d_out;

  // one wave per site: 65536 waves = 8192 blocks x 8 waves
  dim3 block(256);
  dim3 grid(NSITE / 8);
  lptc_wmma_kernel<<<grid, block, 0, stream>>>(x_re, x_im, U_re, U_im,
                                               w_re, w_im, out);
}